// ROIAlign_40261023432732
// MI455X (gfx1250) — compile-verified
//
#include <hip/hip_runtime.h>
#include <hip/hip_bf16.h>
#include <stdint.h>

// ROI Align for MI455X (gfx1250).
// Strategy: per (roi, 4-channel group) block, DMA the ROI's 2D feature-map
// footprint into LDS with the Tensor Data Mover (one tensor_load_to_lds per
// wave32, one channel per wave), then do all bilinear gathers from LDS.
// This replaces ~200M scattered 4B global loads with coalesced 2D-tile DMA
// (input fits in L2: 82MB < 192MB) + LDS gathers.

typedef unsigned int v4u __attribute__((ext_vector_type(4)));
typedef int          v8i __attribute__((ext_vector_type(8)));
typedef int          v4i __attribute__((ext_vector_type(4)));

#define ROI_C   256
#define ROI_H   100
#define ROI_W   100
#define OUT_HW  7
#define RMAX    56          // max tile side (ROI extent <= ~53 by construction)
#define CPB     4           // channels per block == waves per block

__global__ __launch_bounds__(128)
void roi_align_tdm_kernel(const float* __restrict__ input,
                          const float* __restrict__ rois,
                          float* __restrict__ out)
{
    __shared__ float smem[CPB * RMAX * RMAX];   // 50176 B

    const int k    = blockIdx.x;
    const int c0   = blockIdx.y * CPB;
    const int tid  = threadIdx.x;
    const int wave = tid >> 5;       // 0..3 -> channel within group
    const int lane = tid & 31;

    // ---- ROI parameters (uniform per block; compiler scalarizes) ----
    const float* r = rois + (size_t)k * 5;
    const int   b  = (int)r[0];
    const float x1 = r[1] * 0.25f;
    const float y1 = r[2] * 0.25f;
    const float x2 = r[3] * 0.25f;
    const float y2 = r[4] * 0.25f;
    const float roi_w = fmaxf(x2 - x1, 1.0f);
    const float roi_h = fmaxf(y2 - y1, 1.0f);
    const float bin_w = roi_w * (1.0f / OUT_HW);
    const float bin_h = roi_h * (1.0f / OUT_HW);

    // ---- bounding box of all sample corner pixels (uniform) ----
    // samples: g = lo + bin*(p + {0.25,0.75}), p in [0,6] -> min at 0.25, max at 6.75
    const float gx_lo = x1 + bin_w * 0.25f, gx_hi = x1 + bin_w * 6.75f;
    const float gy_lo = y1 + bin_h * 0.25f, gy_hi = y1 + bin_h * 6.75f;
    const int x0i = (int)floorf(fminf(fmaxf(gx_lo, 0.0f), (float)(ROI_W - 1)));
    const int y0i = (int)floorf(fminf(fmaxf(gy_lo, 0.0f), (float)(ROI_H - 1)));
    int xhi = min((int)floorf(fminf(fmaxf(gx_hi, 0.0f), (float)(ROI_W - 1))) + 1, ROI_W - 1);
    int yhi = min((int)floorf(fminf(fmaxf(gy_hi, 0.0f), (float)(ROI_H - 1))) + 1, ROI_H - 1);
    int wid = min(xhi - x0i + 1, RMAX);   // defensive clamp; math bounds <= 54
    int hgt = min(yhi - y0i + 1, RMAX);

    const int ch = c0 + wave;
    const int tile_elems  = wid * hgt;
    const int tile_stride = (tile_elems + 3) & ~3;     // 16B-align channel bases
    float* tile = &smem[wave * tile_stride];

    const size_t gelem = (((size_t)(b * ROI_C + ch)) * ROI_H + (size_t)y0i) * ROI_W
                         + (size_t)x0i;

#if defined(__has_builtin) && __has_builtin(__builtin_amdgcn_tensor_load_to_lds)
    // ---- TDM: one 2D tile DMA per wave (EXEC ignored; per-wave issue) ----
    {
        const unsigned long long gaddr =
            (unsigned long long)(uintptr_t)input + gelem * 4ull;
        const unsigned int lds_off = (unsigned int)(uintptr_t)tile; // low 32b = LDS byte offset

        // D# group 0 (ISA 8.3): count=1 | lds_addr | global_addr[56:0] | type=2
        v4u g0;
        g0.x = 1u;
        g0.y = lds_off;
        g0.z = (unsigned int)(gaddr & 0xFFFFFFFFull);
        g0.w = (unsigned int)((gaddr >> 32) & 0x1FFFFFFull) | (2u << 30);

        // D# group 1 (ISA 8.4): data_size=4B; tensor_dim = tile_dim = (wid,hgt)
        // measured from tile start (region pre-clamped in-bounds -> no OOB);
        // tensor_dim0_stride = W (row pitch of the feature map, in elements).
        v8i g1;
        g1[0] = (int)(2u << 16);                         // wg_mask=0, data_size=2(4B)
        g1[1] = (int)((unsigned int)wid << 16);          // tensor_dim0 lo16 @ [63:48]
        g1[2] = (int)((unsigned int)hgt << 16);          // dim0 hi=0 | tensor_dim1 lo16
        g1[3] = (int)((unsigned int)wid << 16);          // dim1 hi=0 | tile_dim0
        g1[4] = (int)((unsigned int)hgt);                // tile_dim1 | tile_dim2=0
        g1[5] = ROI_W;                                   // tensor_dim0_stride lo32
        g1[6] = 0;                                       // stride hi | dim1_stride lo
        g1[7] = 0;                                       // dim1_stride hi
        const v4i z4 = {0, 0, 0, 0};                     // groups 2/3 unused (2D tile)
        const v8i z8 = {0, 0, 0, 0, 0, 0, 0, 0};         // trailing group (clang-23 form)

        // clang-23 / therock-10.0 signature: (v4u, v8i, v4i, v4i, v8i, i32 cpol)
        __builtin_amdgcn_tensor_load_to_lds(g0, g1, z4, z4, z8, 0);
        __builtin_amdgcn_s_wait_tensorcnt(0);
    }
#else
    // ---- fallback: coalesced wave32 row loads into LDS ----
    for (int ry = 0; ry < hgt; ++ry) {
        const float* grow = input + gelem + (size_t)ry * ROI_W;
        for (int rx = lane; rx < wid; rx += 32)
            tile[ry * wid + rx] = grow[rx];
    }
#endif
    __syncthreads();

    // ---- interpolate: lane -> output bin, wave -> channel ----
    const float fH = (float)ROI_H, fW = (float)ROI_W;
    for (int bin = lane; bin < OUT_HW * OUT_HW; bin += 32) {
        const int ph = bin / OUT_HW;
        const int pw = bin - ph * OUT_HW;
        float acc = 0.0f;
#pragma unroll
        for (int s = 0; s < 4; ++s) {
            const float fy = (s & 2) ? 0.75f : 0.25f;
            const float fx = (s & 1) ? 0.75f : 0.25f;
            const float gy = y1 + bin_h * ((float)ph + fy);
            const float gx = x1 + bin_w * ((float)pw + fx);
            const bool valid = (gy >= -1.0f) && (gy <= fH) && (gx >= -1.0f) && (gx <= fW);

            const float y  = fminf(fmaxf(gy, 0.0f), fH - 1.0f);
            const float x  = fminf(fmaxf(gx, 0.0f), fW - 1.0f);
            const float yl = floorf(y);
            const float xl = floorf(x);
            const float ly = y - yl;
            const float lx = x - xl;

            int iyl = (int)yl, ixl = (int)xl;
            int iyh = min(iyl + 1, ROI_H - 1);
            int ixh = min(ixl + 1, ROI_W - 1);
            // tile-relative (in-range by construction; clamp defensively)
            int ryl = min(max(iyl - y0i, 0), hgt - 1);
            int ryh = min(max(iyh - y0i, 0), hgt - 1);
            int rxl = min(max(ixl - x0i, 0), wid - 1);
            int rxh = min(max(ixh - x0i, 0), wid - 1);

            const float vll = tile[ryl * wid + rxl];
            const float vlh = tile[ryl * wid + rxh];
            const float vhl = tile[ryh * wid + rxl];
            const float vhh = tile[ryh * wid + rxh];

            const float top = (1.0f - lx) * vll + lx * vlh;
            const float bot = (1.0f - lx) * vhl + lx * vhh;
            const float v   = (1.0f - ly) * top + ly * bot;
            acc += valid ? v : 0.0f;
        }
        // out[k][ch][ph][pw], mean over the 2x2 samples
        out[(((size_t)k * ROI_C + ch) * (OUT_HW * OUT_HW)) + bin] = acc * 0.25f;
    }
}

extern "C" void kernel_launch(void* const* d_in, const int* in_sizes, int n_in,
                              void* d_out, int out_size, void* d_ws, size_t ws_size,
                              hipStream_t stream) {
    (void)n_in; (void)out_size; (void)d_ws; (void)ws_size;
    const float* input = (const float*)d_in[0];   // (8,256,100,100) f32
    const float* rois  = (const float*)d_in[1];   // (K,5) f32
    float* out = (float*)d_out;                   // (K,256,7,7) f32

    const int K = in_sizes[1] / 5;                // 1000
    dim3 grid(K, ROI_C / CPB);                    // 1000 x 64 blocks
    roi_align_tdm_kernel<<<grid, 128, 0, stream>>>(input, rois, out);
}